// GCN_26285199851972
// MI455X (gfx1250) — compile-verified
//
#include <hip/hip_runtime.h>

typedef float v2f __attribute__((ext_vector_type(2)));
typedef float v8f __attribute__((ext_vector_type(8)));

#define HID 128
#define NGRAPH 64
#define NCLS 7

// ---------------- degree / dinv ----------------
__global__ void k_deg_init(float* __restrict__ deg, int n) {
  int i = blockIdx.x * 256 + threadIdx.x;
  if (i < n) deg[i] = 1.0f;                       // self-loop
}
__global__ void k_deg_edges(const int* __restrict__ col, float* __restrict__ deg, int e) {
  int i = blockIdx.x * 256 + threadIdx.x;
  if (i < e) atomicAdd(&deg[col[i]], 1.0f);
}
__global__ void k_make_dinv(float* __restrict__ deg, int n) {
  int i = blockIdx.x * 256 + threadIdx.x;
  if (i < n) deg[i] = rsqrtf(deg[i]);             // deg >= 1 always
}

// ---------------- layer 1 (F_IN == 1 -> scalar aggregation) ----------------
__global__ void k_l1_self(const float* __restrict__ x, const float* __restrict__ dinv,
                          float* __restrict__ s, int n) {
  int i = blockIdx.x * 256 + threadIdx.x;
  if (i < n) s[i] = x[i] * dinv[i] * dinv[i];
}
__global__ void k_l1_edges(const float* __restrict__ x, const float* __restrict__ dinv,
                           const int* __restrict__ row, const int* __restrict__ col,
                           float* __restrict__ s, int e) {
  int i = blockIdx.x * 256 + threadIdx.x;
  if (i < e) {
    int r = row[i], c = col[i];
    atomicAdd(&s[c], x[r] * dinv[r] * dinv[c]);
  }
}
__global__ void k_l1_expand(const float* __restrict__ s, const float* __restrict__ W1,
                            const float* __restrict__ b1, float* __restrict__ h1, long total) {
  long idx = (long)blockIdx.x * 256 + threadIdx.x;
  if (idx >= total) return;
  long n = idx >> 7;
  int hc = (int)(idx & 127);
  h1[idx] = fmaxf(fmaf(s[n], W1[hc], b1[hc]), 0.0f);
}

// ---------------- W transpose (WT[c*128+k] = W[k*128+c]) ----------------
__global__ void k_transposeW(const float* __restrict__ W, float* __restrict__ WT) {
  int i = blockIdx.x * 256 + threadIdx.x;      // 0 .. 16383
  int k = i >> 7, c = i & 127;
  WT[c * HID + k] = W[i];
}

// ---------------- fp32 WMMA GEMM: t[n,:] = (h[n,:] @ W) * dinv[n] ----------------
// One wave per 16-row chunk; 8 waves per block. A panel (16x128) cached in v2f a[32]
// per the ISA 16x4 fp32 A layout; B fragments read as float2 from transposed W.
__global__ void __launch_bounds__(256)
k_gemm_scale(const float* __restrict__ h, const float* __restrict__ WT,
             const float* __restrict__ dinv, float* __restrict__ t, int n_nodes) {
  int wave = threadIdx.x >> 5;
  int lane = threadIdx.x & 31;
  long chunk = (long)blockIdx.x * 8 + wave;
  long rowbase = chunk * 16;
  if (rowbase >= n_nodes) return;               // uniform per wave (N % 16 == 0)

  int m    = lane & 15;          // M index (A) / N column (B,D)
  int koff = (lane >> 4) << 1;   // lanes 16-31 hold K = 2,3
  int moff = (lane >> 4) << 3;   // lanes 16-31 hold rows M+8 of C/D

  // A: lane holds h[rowbase+m][kk*4+koff .. +1] for kk = 0..31
  v2f a[32];
  const float* hrow = h + (rowbase + m) * HID;
#pragma unroll
  for (int kk = 0; kk < 32; ++kk) {
    float2 av = *(const float2*)(hrow + kk * 4 + koff);
    a[kk].x = av.x; a[kk].y = av.y;
  }
  float dv[8];
#pragma unroll
  for (int r = 0; r < 8; ++r) dv[r] = dinv[rowbase + r + moff];

  for (int j = 0; j < 8; ++j) {                 // 8 column tiles of 16
    v8f acc = {};
    const float* wt = WT + (j * 16 + m) * HID;  // column j*16+m of W, K contiguous
#pragma unroll
    for (int kk = 0; kk < 32; ++kk) {
      float2 bv = *(const float2*)(wt + kk * 4 + koff);
      v2f b; b.x = bv.x; b.y = bv.y;
      acc = __builtin_amdgcn_wmma_f32_16x16x4_f32(
          false, a[kk], false, b, (short)0, acc, false, false);
    }
#pragma unroll
    for (int r = 0; r < 8; ++r) {
      t[(rowbase + r + moff) * HID + j * 16 + m] = acc[r] * dv[r];
    }
  }
}

// ---------------- edge scatter: acc[col] += t[row] (t already scaled by dinv[row]) ----------------
__global__ void k_scatter(const float* __restrict__ t, const int* __restrict__ row,
                          const int* __restrict__ col, float* __restrict__ acc, long total) {
  long idx = (long)blockIdx.x * 256 + threadIdx.x;
  if (idx >= total) return;
  long e = idx >> 7;
  int hc = (int)(idx & 127);
  int r = row[e], c = col[e];
  atomicAdd(acc + (long)c * HID + hc, t[(long)r * HID + hc]);
}

// ---------------- epilogue: h = [relu](acc * dinv[n] + b) ----------------
__global__ void k_epilogue(float* __restrict__ acc, const float* __restrict__ dinv,
                           const float* __restrict__ b, int relu, long total) {
  long idx = (long)blockIdx.x * 256 + threadIdx.x;
  if (idx >= total) return;
  long n = idx >> 7;
  int hc = (int)(idx & 127);
  float v = fmaf(acc[idx], dinv[n], b[hc]);
  if (relu) v = fmaxf(v, 0.0f);
  acc[idx] = v;
}

// ---------------- pooling ----------------
__global__ void k_zero(float* __restrict__ p, int n) {
  int i = blockIdx.x * 256 + threadIdx.x;
  if (i < n) p[i] = 0.0f;
}
__global__ void k_count(const int* __restrict__ batch, float* __restrict__ cnt, int n) {
  int i = blockIdx.x * 256 + threadIdx.x;
  if (i < n) atomicAdd(&cnt[batch[i]], 1.0f);
}
// batch is sorted: running per-column sums, atomics only at graph boundaries
__global__ void __launch_bounds__(128)
k_pool_sorted(const float* __restrict__ h3, const int* __restrict__ batch,
              float* __restrict__ pooled, int n_nodes, int chunk) {
  int hc = threadIdx.x;                         // 0..127
  int n0 = blockIdx.x * chunk;
  if (n0 >= n_nodes) return;
  int n1 = n0 + chunk; if (n1 > n_nodes) n1 = n_nodes;
  int g = batch[n0];
  float accum = 0.0f;
  for (int n = n0; n < n1; ++n) {
    int b = batch[n];
    if (b != g) {
      atomicAdd(&pooled[(long)g * HID + hc], accum);
      accum = 0.0f; g = b;
    }
    accum += h3[(long)n * HID + hc];
  }
  atomicAdd(&pooled[(long)g * HID + hc], accum);
}

// ---------------- head: out[g,c] = (pooled[g,:]/cnt[g]) @ Wl + bl ----------------
__global__ void k_head(const float* __restrict__ pooled, const float* __restrict__ cnt,
                       const float* __restrict__ Wl, const float* __restrict__ bl,
                       float* __restrict__ out) {
  int i = blockIdx.x * 256 + threadIdx.x;
  if (i >= NGRAPH * NCLS) return;
  int g = i / NCLS, c = i % NCLS;
  float inv = 1.0f / fmaxf(cnt[g], 1.0f);
  float s = 0.0f;
  for (int hh = 0; hh < HID; ++hh) s += pooled[(long)g * HID + hh] * Wl[hh * NCLS + c];
  out[i] = fmaf(s, inv, bl[c]);
}

extern "C" void kernel_launch(void* const* d_in, const int* in_sizes, int n_in,
                              void* d_out, int out_size, void* d_ws, size_t ws_size,
                              hipStream_t stream) {
  const float* x     = (const float*)d_in[0];
  const int*   ei    = (const int*)d_in[1];
  const int*   batch = (const int*)d_in[2];
  const float* W1    = (const float*)d_in[3];
  const float* b1    = (const float*)d_in[4];
  const float* W2    = (const float*)d_in[5];
  const float* b2    = (const float*)d_in[6];
  const float* W3    = (const float*)d_in[7];
  const float* b3    = (const float*)d_in[8];
  const float* Wl    = (const float*)d_in[9];
  const float* bl    = (const float*)d_in[10];
  float* out = (float*)d_out;

  const int N = in_sizes[0];            // F_IN == 1
  const int E = in_sizes[1] / 2;
  const int* row = ei;                  // edge_index[0] = source
  const int* col = ei + E;              // edge_index[1] = target

  // workspace carve-up (all fp32)
  float* dinv   = (float*)d_ws;                 // N
  float* s      = dinv + N;                     // N
  float* WT     = s + N;                        // 128*128
  float* cnt    = WT + HID * HID;               // 64
  float* pooled = cnt + NGRAPH;                 // 64*128
  float* bufA   = pooled + NGRAPH * HID;        // N*128 (h / accumulator)
  float* bufB   = bufA + (size_t)N * HID;       // N*128 (t = h@W * dinv)

  const long totNH = (long)N * HID;
  const long totEH = (long)E * HID;
  dim3 blk(256);
  int gN   = (N + 255) / 256;
  int gE   = (E + 255) / 256;
  int gNH  = (int)((totNH + 255) / 256);
  int gEH  = (int)((totEH + 255) / 256);
  int gemmBlocks = (N / 16 + 7) / 8;
  int poolChunk = 256;
  int gPool = (N + poolChunk - 1) / poolChunk;

  // degrees -> dinv
  k_deg_init<<<gN, blk, 0, stream>>>(dinv, N);
  k_deg_edges<<<gE, blk, 0, stream>>>(col, dinv, E);
  k_make_dinv<<<gN, blk, 0, stream>>>(dinv, N);

  // zero pooling buffers (cnt and pooled are contiguous)
  k_zero<<<(NGRAPH + NGRAPH * HID + 255) / 256, blk, 0, stream>>>(cnt, NGRAPH + NGRAPH * HID);

  // layer 1 (scalar aggregation, then broadcast through W1)
  k_l1_self<<<gN, blk, 0, stream>>>(x, dinv, s, N);
  k_l1_edges<<<gE, blk, 0, stream>>>(x, dinv, row, col, s, E);
  k_l1_expand<<<gNH, blk, 0, stream>>>(s, W1, b1, bufA, totNH);

  // layer 2: t = (h1@W2)*dinv ; acc = t(self) + scatter ; h2 = relu(acc*dinv + b2)
  k_transposeW<<<HID * HID / 256, blk, 0, stream>>>(W2, WT);
  k_gemm_scale<<<gemmBlocks, blk, 0, stream>>>(bufA, WT, dinv, bufB, N);
  hipMemcpyAsync(bufA, bufB, totNH * sizeof(float), hipMemcpyDeviceToDevice, stream);
  k_scatter<<<gEH, blk, 0, stream>>>(bufB, row, col, bufA, totEH);
  k_epilogue<<<gNH, blk, 0, stream>>>(bufA, dinv, b2, 1, totNH);

  // layer 3 (no relu)
  k_transposeW<<<HID * HID / 256, blk, 0, stream>>>(W3, WT);
  k_gemm_scale<<<gemmBlocks, blk, 0, stream>>>(bufA, WT, dinv, bufB, N);
  hipMemcpyAsync(bufA, bufB, totNH * sizeof(float), hipMemcpyDeviceToDevice, stream);
  k_scatter<<<gEH, blk, 0, stream>>>(bufB, row, col, bufA, totEH);
  k_epilogue<<<gNH, blk, 0, stream>>>(bufA, dinv, b3, 0, totNH);

  // mean pool + head
  k_count<<<gN, blk, 0, stream>>>(batch, cnt, N);
  k_pool_sorted<<<gPool, dim3(128), 0, stream>>>(bufA, batch, pooled, N, poolChunk);
  k_head<<<(NGRAPH * NCLS + 255) / 256, blk, 0, stream>>>(pooled, cnt, Wl, bl, out);
}